// postprcess_45698452029741
// MI455X (gfx1250) — compile-verified
//
#include <hip/hip_runtime.h>
#include <hip/hip_bf16.h>

#define THREADS 256
#define ROWS_PER_BLOCK 256
#define ASYNC_TRY 1

typedef __attribute__((ext_vector_type(4))) float v4f;
typedef __attribute__((ext_vector_type(2))) float v2f;
typedef __attribute__((ext_vector_type(4))) int   v4i;

typedef __attribute__((address_space(1))) v4i* gptr_v4i;
typedef __attribute__((address_space(3))) v4i* lptr_v4i;

__device__ __forceinline__ float wrap_pi(float a) {
  const float PI_F  = 3.14159265358979323846f;
  const float TPI_F = 6.28318530717958647692f;
  a = (a >  PI_F) ? a - TPI_F : a;
  a = (a < -PI_F) ? a + TPI_F : a;
  return a;
}

__global__ __launch_bounds__(THREADS) void postprocess_kernel(
    const float* __restrict__ reg,      // [n, 50]
    const float* __restrict__ centers,  // [n, 2]
    const int*   __restrict__ clses,    // [n]
    float* __restrict__ out,            // [n, 16]
    int n)
{
  __shared__ float s_reg[ROWS_PER_BLOCK * 50];   // 51200 B, reused for output staging

  const int tid  = threadIdx.x;
  const long long row0 = (long long)blockIdx.x * ROWS_PER_BLOCK;
  int rows = n - (int)row0;
  if (rows <= 0) return;
  if (rows > ROWS_PER_BLOCK) rows = ROWS_PER_BLOCK;

  const float* gsrc = reg + row0 * 50;
  const int ndw  = rows * 50;
  const int nvec = ndw >> 2;     // block-base is 51200B-aligned -> 16B OK

  // ---- stage input tile into LDS with coalesced 16B transfers ----
#if ASYNC_TRY && __has_builtin(__builtin_amdgcn_global_load_async_to_lds_b128)
  for (int i = tid; i < nvec; i += THREADS) {
    __builtin_amdgcn_global_load_async_to_lds_b128(
        (gptr_v4i)(gsrc + 4 * i),
        (lptr_v4i)(&s_reg[4 * i]),
        0, 0);
  }
  #if __has_builtin(__builtin_amdgcn_s_wait_asynccnt)
  __builtin_amdgcn_s_wait_asynccnt(0);
  #else
  asm volatile("s_wait_asynccnt 0" ::: "memory");
  #endif
#else
  for (int i = tid; i < nvec; i += THREADS) {
    v4f v = __builtin_nontemporal_load((const v4f*)gsrc + i);
    *((v4f*)s_reg + i) = v;
  }
#endif
  // scalar remainder (none for rows in {128,256}, kept for generality)
  for (int i = (nvec << 2) + tid; i < ndw; i += THREADS) s_reg[i] = gsrc[i];
  __syncthreads();

  // ---- per-row decode ----
  float o[16];
  if (tid < rows) {
    const long long r = row0 + tid;
    const float* R = &s_reg[tid * 50];
    const v2f c = *((const v2f*)centers + r);
    const float cx = c.x, cy = c.y;
    const int cls = clses[r];

    // box2d: cols 0,1 clipped (col1 twice -> [0,384]); cols 2,3 NOT clipped (ref quirk)
    float x1 = (cx - R[0]) * 4.0f;
    float y1 = (cy - R[1]) * 4.0f;
    float x2 = (cx + R[2]) * 4.0f;
    float y2 = (cy + R[3]) * 4.0f;
    o[0] = fminf(fmaxf(x1, 0.0f), 1280.0f);
    o[1] = fminf(fmaxf(y1, 0.0f), 384.0f);
    o[2] = x2;
    o[3] = y2;

    // dims = exp(dim3d) * DIM_MEAN[cls]
    float dmx = (cls == 0) ? 4.83899871f : ((cls == 1) ? 0.91986743f : 1.78652745f);
    float dmy = (cls == 0) ? 1.80778956f : ((cls == 1) ? 1.75302337f : 1.76500989f);
    float dmz = (cls == 0) ? 2.11565798f : ((cls == 1) ? 0.86220807f : 0.83395625f);
    float d0 = expf(R[29]) * dmx;
    float d1 = expf(R[30]) * dmy;   // h3d
    float d2 = expf(R[31]) * dmz;
    o[4] = d0; o[5] = d1; o[6] = d2;

    // depth: 1/sigmoid(x)-1 == exp(-x), clipped
    float depth = fminf(fmaxf(expf(-R[48]), 0.1f), 100.0f);
    o[7] = depth;

    // keypoint depths; kp[:,k,1] == R[7+2k]
    const float F_U = 721.5377f;
    float fh = F_U * d1;
    float cen  = fh / (fmaxf(R[23] - R[25], 0.0f) * 4.0f + 1e-3f);
    float c02a = fh / (fmaxf(R[7]  - R[15], 0.0f) * 4.0f + 1e-3f);
    float c02b = fh / (fmaxf(R[11] - R[19], 0.0f) * 4.0f + 1e-3f);
    float c13a = fh / (fmaxf(R[9]  - R[17], 0.0f) * 4.0f + 1e-3f);
    float c13b = fh / (fmaxf(R[13] - R[21], 0.0f) * 4.0f + 1e-3f);
    o[8]  = fminf(fmaxf(cen,                 0.1f), 100.0f);
    o[9]  = fminf(fmaxf(0.5f * (c02a + c02b), 0.1f), 100.0f);
    o[10] = fminf(fmaxf(0.5f * (c13a + c13b), 0.1f), 100.0f);

    // locations
    float px = (cx + R[4]) * 4.0f;
    float py = (cy + R[5]) * 4.0f;
    float loc_x = (px - 609.5593f) * depth / F_U;
    float loc_y = (py - 172.854f)  * depth / F_U;
    o[11] = loc_x; o[12] = loc_y; o[13] = depth;

    // orientation: softmax prob[1] == sigmoid(b-a) -> argmax of (b-a), first-max ties
    float best = R[33] - R[32];
    int bin = 0;
    { float s = R[35] - R[34]; if (s > best) { best = s; bin = 1; } }
    { float s = R[37] - R[36]; if (s > best) { best = s; bin = 2; } }
    { float s = R[39] - R[38]; if (s > best) { best = s; bin = 3; } }
    float off0 = R[40 + 2 * bin];
    float off1 = R[41 + 2 * bin];
    float ac = (bin == 0) ? 0.0f
             : (bin == 1) ? 1.57079632679489662f
             : (bin == 2) ? 3.14159274101257324f
                          : -1.57079632679489662f;
    float alpha = atanf(off0 / off1) + ac;     // atan of quotient, as in reference
    float ray   = atanf(loc_x / depth);
    o[14] = wrap_pi(alpha + ray);              // roty uses UNwrapped alpha
    o[15] = wrap_pi(alpha);
  }
  __syncthreads();   // all s_reg reads complete before reuse

  // ---- stage outputs (stride-17 padding: conflict-free) then NT streaming stores ----
  float* s_out = s_reg;
  if (tid < rows) {
#pragma unroll
    for (int cc = 0; cc < 16; ++cc) s_out[tid * 17 + cc] = o[cc];
  }
  __syncthreads();

  float* gdst = out + row0 * 16;
  const int odw4 = (rows * 16) >> 2;           // rows*16 always %4 == 0
  for (int i = tid; i < odw4; i += THREADS) {
    int base = i << 2;
    int rw = base >> 4;
    int cc = base & 12;
    v4f v;
    v.x = s_out[rw * 17 + cc + 0];
    v.y = s_out[rw * 17 + cc + 1];
    v.z = s_out[rw * 17 + cc + 2];
    v.w = s_out[rw * 17 + cc + 3];
    __builtin_nontemporal_store(v, (v4f*)gdst + i);
  }
}

extern "C" void kernel_launch(void* const* d_in, const int* in_sizes, int n_in,
                              void* d_out, int out_size, void* d_ws, size_t ws_size,
                              hipStream_t stream) {
  const float* reg     = (const float*)d_in[0];   // [N,50] f32
  const float* centers = (const float*)d_in[1];   // [N,2]  f32
  const int*   clses   = (const int*)d_in[2];     // [N]    int
  float* out = (float*)d_out;                     // [N,16] f32

  const int n = in_sizes[2];                      // clses element count == N
  const int grid = (n + ROWS_PER_BLOCK - 1) / ROWS_PER_BLOCK;
  if (grid > 0) {
    postprocess_kernel<<<grid, THREADS, 0, stream>>>(reg, centers, clses, out, n);
  }
}